// GNNModel_57071525429602
// MI455X (gfx1250) — compile-verified
//
#include <hip/hip_runtime.h>
#include <hip/hip_bf16.h>

typedef __attribute__((ext_vector_type(2))) float v2f;
typedef __attribute__((ext_vector_type(8))) float v8f;

#define DIM 128
#define LDSW 132   // padded LDS row stride (floats): bank stride 4 mod 64 -> conflict-free b64 reads

// ---------------------------------------------------------------------------
// Zero the aggregation buffer (float4 stores).
// ---------------------------------------------------------------------------
__global__ __launch_bounds__(256) void zero_kernel(float4* __restrict__ p, int n4) {
    int i = blockIdx.x * blockDim.x + threadIdx.x;
    if (i < n4) p[i] = make_float4(0.f, 0.f, 0.f, 0.f);
}

// ---------------------------------------------------------------------------
// Edge scatter-add: agg[dst] += h[src].  One wave (32 lanes) per edge,
// each lane handles 4 consecutive floats (float4 gather + 4 f32 atomics).
// Feature matrix (25.6 MB) is L2-resident, so atomics resolve in GL2.
// ---------------------------------------------------------------------------
__global__ __launch_bounds__(256) void scatter_add_kernel(
    const float* __restrict__ h,
    const int*   __restrict__ src,
    const int*   __restrict__ dst,
    float*       __restrict__ agg,
    int num_edges)
{
    int tid  = blockIdx.x * blockDim.x + threadIdx.x;
    int edge = tid >> 5;
    int lane = tid & 31;
    if (edge >= num_edges) return;

    int s = src[edge];
    int d = dst[edge];

    const float4* xs = (const float4*)(h + (size_t)s * DIM);
    float4 v = xs[lane];

    float* ad = agg + (size_t)d * DIM + lane * 4;
    atomicAdd(ad + 0, v.x);
    atomicAdd(ad + 1, v.y);
    atomicAdd(ad + 2, v.z);
    atomicAdd(ad + 3, v.w);
}

// ---------------------------------------------------------------------------
// Fused GraphConv linear stage:
//   out = act( agg @ W_rel + bias + hin @ W_root )
//
// One wave computes a full 16-row x 128-col output stripe (8 column tiles),
// so A fragments (agg / hin rows) are loaded from global exactly once and
// amortized over 16 WMMAs per k-step.  Both 128x128 weight matrices are
// staged transposed in dynamic LDS (2 x 128 x 132 floats = 132 KB) with a
// padded stride of 132 so each wave's ds_load_b64 hits all 64 banks.
//
// The 16 per-tile LDS base pointers are hoisted out of the (fully unrolled)
// k-loop so every DS/global access is base + compile-time immediate: the
// steady-state instruction stream is pure load/wmma/wait with no per-step
// address VALU.
//
// WMMA f32 16x16x4 fragment mapping (per ISA 7.12.2):
//   A (16x4): lane L -> M = L%16, VGPR0/1 hold K = 2*(L/16) + {0,1}
//   B (4x16): lane L -> N = L%16, VGPR0/1 hold K = 2*(L/16) + {0,1}
//   C (16x16): VGPR r -> M = r + 8*(L/16), N = L%16
// Both fragments are therefore a single aligned float2 load.
// ---------------------------------------------------------------------------
template <bool RELU>
__global__ __launch_bounds__(256) void gemm_bias_root_kernel(
    const float* __restrict__ agg,    // [N,D] aggregated neighbors
    const float* __restrict__ hin,    // [N,D] root features
    const float* __restrict__ Wrel,   // [D,D]
    const float* __restrict__ Wroot,  // [D,D]
    const float* __restrict__ bias,   // [D]
    float*       __restrict__ out,    // [N,D]
    int n_rows)
{
    extern __shared__ float lds[];
    float* lw_rel  = lds;               // [DIM][LDSW] transposed: lw[n][k]
    float* lw_root = lds + DIM * LDSW;

    // Cooperative transposed load of both 128x128 weight matrices.
    // Consecutive threads read consecutive n (coalesced global b32 reads).
    for (int i = threadIdx.x; i < DIM * DIM; i += 256) {
        int k = i >> 7;        // row of W   (K index)
        int n = i & (DIM - 1); // col of W   (N index)
        lw_rel [n * LDSW + k] = Wrel [i];
        lw_root[n * LDSW + k] = Wroot[i];
    }
    __syncthreads();

    const int wave = threadIdx.x >> 5;
    const int lane = threadIdx.x & 31;
    const int rowBase = (blockIdx.x * 8 + wave) * 16;

    // Wave-uniform bounds check (keeps EXEC all-ones for WMMA).
    if (__builtin_amdgcn_readfirstlane(rowBase) < n_rows) {
        const int m  = lane & 15;        // M (for A) / N (for B, C)
        const int kh = (lane >> 4) * 2;  // K sub-offset: 0 or 2

        const float* arow = agg + (size_t)(rowBase + m) * DIM + kh;
        const float* xrow = hin + (size_t)(rowBase + m) * DIM + kh;

        // Hoisted per-tile LDS fragment base pointers (constant inside loop).
        const float* brel[8];
        const float* brt [8];
#pragma unroll
        for (int t = 0; t < 8; ++t) {
            brel[t] = lw_rel  + (t * 16 + m) * LDSW + kh;
            brt [t] = lw_root + (t * 16 + m) * LDSW + kh;
        }

        v8f acc[8];
#pragma unroll
        for (int t = 0; t < 8; ++t) acc[t] = (v8f){};

#pragma unroll
        for (int k = 0; k < DIM; k += 4) {           // fully unrolled: all
            v2f a0 = *(const v2f*)(arow + k);        // offsets are immediates
            v2f a1 = *(const v2f*)(xrow + k);
#pragma unroll
            for (int t = 0; t < 8; ++t) {
                v2f b0 = *(const v2f*)(brel[t] + k);
                acc[t] = __builtin_amdgcn_wmma_f32_16x16x4_f32(
                             false, a0, false, b0, (short)0, acc[t], false, false);
                v2f b1 = *(const v2f*)(brt[t] + k);
                acc[t] = __builtin_amdgcn_wmma_f32_16x16x4_f32(
                             false, a1, false, b1, (short)0, acc[t], false, false);
            }
        }

        const int mBase = rowBase + ((lane >> 4) * 8);
#pragma unroll
        for (int t = 0; t < 8; ++t) {
            const float bv = bias[t * 16 + m];
#pragma unroll
            for (int r = 0; r < 8; ++r) {
                float v = acc[t][r] + bv;
                if (RELU) v = fmaxf(v, 0.0f);
                out[(size_t)(mBase + r) * DIM + t * 16 + m] = v;
            }
        }
    }
}

// ---------------------------------------------------------------------------
// Launch: two GraphConv layers.
//   ws layout: agg [N*D floats] | h [N*D floats]   (51.2 MB total)
// ---------------------------------------------------------------------------
extern "C" void kernel_launch(void* const* d_in, const int* in_sizes, int n_in,
                              void* d_out, int out_size, void* d_ws, size_t ws_size,
                              hipStream_t stream) {
    const int N = in_sizes[0] / DIM;      // 50000
    const int E = in_sizes[1] / 2;        // 625000

    const float* x       = (const float*)d_in[0];
    const int*   ei      = (const int*)d_in[1];
    const float* W1_rel  = (const float*)d_in[2];
    const float* b1_rel  = (const float*)d_in[3];
    const float* W1_root = (const float*)d_in[4];
    const float* W2_rel  = (const float*)d_in[5];
    const float* b2_rel  = (const float*)d_in[6];
    const float* W2_root = (const float*)d_in[7];

    const int* src = ei;
    const int* dst = ei + E;

    float* agg = (float*)d_ws;
    float* h   = agg + (size_t)N * DIM;
    float* out = (float*)d_out;

    const int n4 = N * DIM / 4;
    const dim3 zeroGrid((n4 + 255) / 256);
    const dim3 scatGrid((E * 32 + 255) / 256);
    const dim3 gemmGrid((N / 16 + 7) / 8);
    const dim3 block(256);
    const size_t shmem = (size_t)2 * DIM * LDSW * sizeof(float);  // 132 KB

    // ---- Layer 1: h = relu(agg(x) @ W1_rel + b1 + x @ W1_root) ----
    zero_kernel<<<zeroGrid, block, 0, stream>>>((float4*)agg, n4);
    scatter_add_kernel<<<scatGrid, block, 0, stream>>>(x, src, dst, agg, E);
    gemm_bias_root_kernel<true><<<gemmGrid, block, shmem, stream>>>(
        agg, x, W1_rel, W1_root, b1_rel, h, N);

    // ---- Layer 2: out = agg(h) @ W2_rel + b2 + h @ W2_root ----
    zero_kernel<<<zeroGrid, block, 0, stream>>>((float4*)agg, n4);
    scatter_add_kernel<<<scatGrid, block, 0, stream>>>(h, src, dst, agg, E);
    gemm_bias_root_kernel<false><<<gemmGrid, block, shmem, stream>>>(
        agg, h, W2_rel, W2_root, b2_rel, out, N);
}